// MicroGCN_39642548142524
// MI455X (gfx1250) — compile-verified
//
#include <hip/hip_runtime.h>
#include <hip/hip_bf16.h>

typedef __attribute__((ext_vector_type(2))) float v2f;
typedef __attribute__((ext_vector_type(8))) float v8f;

#define NNODES  100000
#define NEDGES  1600000
#define NSTATES 64
#define INDIM   128
#define HIDDIM  128
#define OUTDIM  64

// ---------------- utility kernels ----------------

__global__ void k_fill(float* __restrict__ p, float v, int n) {
    int i = blockIdx.x * blockDim.x + threadIdx.x;
    if (i < n) p[i] = v;
}

__global__ void k_deg(const int* __restrict__ dst, const float* __restrict__ w,
                      float* __restrict__ deg, int E) {
    int e = blockIdx.x * blockDim.x + threadIdx.x;
    if (e < E) atomicAdd(&deg[dst[e]], w[e]);
}

__global__ void k_rsqrt_inplace(float* __restrict__ deg, int n) {
    int i = blockIdx.x * blockDim.x + threadIdx.x;
    if (i < n) deg[i] = rsqrtf(deg[i]);   // deg >= 1 (self-loop), safe
}

__global__ void k_norm(const int* __restrict__ src, const int* __restrict__ dst,
                       const float* __restrict__ w, const float* __restrict__ dinv,
                       float* __restrict__ norm, int E) {
    int e = blockIdx.x * blockDim.x + threadIdx.x;
    if (e < E) norm[e] = dinv[src[e]] * w[e] * dinv[dst[e]];
}

// ---------------- f32 WMMA GEMM:  C[M x Nc] = A[M x K] @ W[K x Nc] ----------------
// One wave32 computes one 16x16 output tile with V_WMMA_F32_16X16X4_F32 (exact f32).
// W is staged in LDS pair-interleaved: the two k-values a lane needs for its B
// fragment sit at adjacent dwords -> single ds_load_b64 per WMMA. Pair-row stride
// is padded by +32 dwords so the two half-waves (k-offset 0 vs 2) hit disjoint
// LDS banks. K and Nc are template constants -> full unroll, immediate offsets.

template<int K, int Nc>
__global__ __launch_bounds__(256)
void k_gemm_wmma_f32(const float* __restrict__ A, const float* __restrict__ W,
                     float* __restrict__ C, int M) {
    constexpr int PSTRIDE = 2 * Nc + 32;          // dwords per k-pair row (+32 = bank shift)
    __shared__ float sW[(K / 2) * PSTRIDE];

    const int tid = threadIdx.x;

    // cooperative stage of W (K x Nc) into pair-interleaved LDS
    for (int idx = tid; idx < K * Nc; idx += 256) {
        int k = idx / Nc;                          // compile-time Nc -> shifts
        int n = idx - k * Nc;
        sW[(k >> 1) * PSTRIDE + n * 2 + (k & 1)] = W[idx];
    }
    __syncthreads();

    const int wave = tid >> 5;                    // 8 waves / block
    const int lane = tid & 31;
    constexpr int nct = Nc >> 4;                  // col tiles (8 or 4)
    const int tile = blockIdx.x * 8 + wave;
    const int m0 = (tile / nct) << 4;
    const int n0 = (tile & (nct - 1)) << 4;
    if (m0 >= M) return;                          // uniform per wave: EXEC all-1s

    // A fragment: lane l holds row m0+(l&15), K pair base koff = (l>>4)*2
    const int half = lane >> 4;                   // 0 or 1
    const int koff = half << 1;                   // 0 or 2
    const int ncol = n0 + (lane & 15);
    const float* Arow = A + (size_t)(m0 + (lane & 15)) * K;
    const float* Bbase = &sW[half * PSTRIDE + ncol * 2]; // + (k0/2)*PSTRIDE per step

    v8f acc = {};
#pragma unroll
    for (int k0 = 0; k0 < K; k0 += 4) {
        v2f a = *(const v2f*)(Arow + k0 + koff);                 // global_load_b64
        v2f b = *(const v2f*)(Bbase + (k0 >> 1) * PSTRIDE);      // ds_load_b64
        acc = __builtin_amdgcn_wmma_f32_16x16x4_f32(
            /*neg_a=*/false, a, /*neg_b=*/false, b,
            /*c_mod=*/(short)0, acc, /*reuse_a=*/false, /*reuse_b=*/false);
    }

    // C/D layout: VGPR v -> row m0 + 8*(lane>>4) + v, col = n0 + (lane&15)
    const int mbase = m0 + (half << 3);
#pragma unroll
    for (int v = 0; v < 8; ++v)
        C[(size_t)(mbase + v) * Nc + ncol] = acc[v];
}

// ---------------- edge scatter-add (gather h[src]*norm, atomic add into agg[dst]) ----
// LANES threads per edge, each thread owns one float4 of the feature vector.
// Accumulator matrix (<=51.2MB) is L2-resident on MI455X (192MB L2), so the
// global_atomic_add_f32 traffic never leaves the chip.

template<int F, int LANES>
__global__ __launch_bounds__(256)
void k_scatter(const float* __restrict__ h,
               const int* __restrict__ src, const int* __restrict__ dst,
               const float* __restrict__ norm, const float* __restrict__ dinv,
               float* __restrict__ agg, int E, int N) {
    long long t = (long long)blockIdx.x * blockDim.x + threadIdx.x;
    long long e = t / LANES;
    int sub = (int)(t - e * LANES);
    if (e >= (long long)E + N) return;

    int s, d; float nm;
    if (e < E) {
        s = src[e]; d = dst[e]; nm = norm[e];
    } else {                       // self-loop: weight 1, norm = dinv^2
        s = d = (int)(e - E);
        float di = dinv[s];
        nm = di * di;
    }
    const float4 hv = *(const float4*)(h + (size_t)s * F + sub * 4);
    float* ap = agg + (size_t)d * F + sub * 4;
    atomicAdd(ap + 0, hv.x * nm);
    atomicAdd(ap + 1, hv.y * nm);
    atomicAdd(ap + 2, hv.z * nm);
    atomicAdd(ap + 3, hv.w * nm);
}

__global__ void k_bias_relu(float* __restrict__ h, const float* __restrict__ b,
                            int n, int F) {
    int i = blockIdx.x * blockDim.x + threadIdx.x;
    if (i < n) {
        float v = h[i] + b[i & (F - 1)];          // F is a power of two
        h[i] = v > 0.f ? v : 0.f;
    }
}

// ---------------- segment-mean pooling ----------------

__global__ void k_pool(const float* __restrict__ h2, const int* __restrict__ state,
                       float* __restrict__ pooled, int N) {
    int t = blockIdx.x * blockDim.x + threadIdx.x;
    if (t < N * OUTDIM) {
        int n = t >> 6;                            // OUTDIM == 64
        int f = t & 63;
        atomicAdd(&pooled[state[n] * OUTDIM + f], h2[t]);
    }
}

__global__ void k_count(const int* __restrict__ state, float* __restrict__ counts, int N) {
    int i = blockIdx.x * blockDim.x + threadIdx.x;
    if (i < N) atomicAdd(&counts[state[i]], 1.0f);
}

__global__ void k_final(const float* __restrict__ pooled, const float* __restrict__ counts,
                        const float* __restrict__ b2, float* __restrict__ out) {
    int t = blockIdx.x * blockDim.x + threadIdx.x;
    if (t < NSTATES * OUTDIM) {
        int s = t >> 6;
        int f = t & 63;
        float c = counts[s];
        c = c > 1.f ? c : 1.f;
        out[t] = pooled[t] / c + b2[f];   // mean(agg2 + b2) == mean(agg2) + b2
    }
}

// ---------------- launcher ----------------

static inline int cdiv(long long a, int b) { return (int)((a + b - 1) / b); }

extern "C" void kernel_launch(void* const* d_in, const int* in_sizes, int n_in,
                              void* d_out, int out_size, void* d_ws, size_t ws_size,
                              hipStream_t stream) {
    const float* x   = (const float*)d_in[0];   // [N,128]
    const int* esrc  = (const int*)  d_in[1];   // [E]
    const int* edst  = (const int*)  d_in[2];   // [E]
    const float* ew  = (const float*)d_in[3];   // [E]
    const int* state = (const int*)  d_in[4];   // [N]
    const float* W1  = (const float*)d_in[5];   // [128,128]
    const float* b1  = (const float*)d_in[6];   // [128]
    const float* W2  = (const float*)d_in[7];   // [128,64]
    const float* b2  = (const float*)d_in[8];   // [64]
    float* out = (float*)d_out;                 // [64,64]

    const int N = NNODES, E = NEDGES;

    // workspace layout (floats)
    float* h0     = (float*)d_ws;               // N*128  (h1 of layer1, later h2)
    float* h1     = h0 + (size_t)N * 128;       // N*128  (agg1, later agg2)
    float* deg    = h1 + (size_t)N * 128;       // N      (deg -> dinv)
    float* norm   = deg + N;                    // E
    float* pooled = norm + E;                   // 64*64
    float* counts = pooled + NSTATES * OUTDIM;  // 64

    // init accumulators (re-zero every call: replay-safe)
    k_fill<<<cdiv(N, 256), 256, 0, stream>>>(deg, 1.0f, N);              // self-loop wt
    k_fill<<<cdiv((long long)N * 128, 256), 256, 0, stream>>>(h1, 0.f, N * 128);
    k_fill<<<cdiv(NSTATES * OUTDIM + NSTATES, 256), 256, 0, stream>>>(pooled, 0.f,
                                                    NSTATES * OUTDIM + NSTATES);

    // normalization coefficients
    k_deg<<<cdiv(E, 256), 256, 0, stream>>>(edst, ew, deg, E);
    k_rsqrt_inplace<<<cdiv(N, 256), 256, 0, stream>>>(deg, N);           // deg := dinv
    k_norm<<<cdiv(E, 256), 256, 0, stream>>>(esrc, edst, ew, deg, norm, E);

    // layer 1: h0 = x @ W1   (50000 tiles, 8 waves/block -> 6250 blocks)
    {
        int tiles = (N / 16) * (HIDDIM / 16);
        k_gemm_wmma_f32<INDIM, HIDDIM><<<tiles / 8, 256, 0, stream>>>(x, W1, h0, N);
    }
    // aggregate into h1, then bias+relu in place
    k_scatter<128, 32><<<cdiv((long long)(E + N) * 32, 256), 256, 0, stream>>>(
        h0, esrc, edst, norm, deg, h1, E, N);
    k_bias_relu<<<cdiv((long long)N * 128, 256), 256, 0, stream>>>(h1, b1, N * 128, 128);

    // layer 2: h0 = h1 @ W2  (25000 tiles -> 3125 blocks)
    {
        int tiles = (N / 16) * (OUTDIM / 16);
        k_gemm_wmma_f32<HIDDIM, OUTDIM><<<tiles / 8, 256, 0, stream>>>(h1, W2, h0, N);
    }
    // aggregate into h1 (reused, first N*64 floats)
    k_fill<<<cdiv((long long)N * 64, 256), 256, 0, stream>>>(h1, 0.f, N * 64);
    k_scatter<64, 16><<<cdiv((long long)(E + N) * 16, 256), 256, 0, stream>>>(
        h0, esrc, edst, norm, deg, h1, E, N);

    // segment-mean pooling by state
    k_pool<<<cdiv((long long)N * OUTDIM, 256), 256, 0, stream>>>(h1, state, pooled, N);
    k_count<<<cdiv(N, 256), 256, 0, stream>>>(state, counts, N);
    k_final<<<cdiv(NSTATES * OUTDIM, 256), 256, 0, stream>>>(pooled, counts, b2, out);
}